// GateCircuit_65481071402958
// MI455X (gfx1250) — compile-verified
//
#include <hip/hip_runtime.h>

// out[b] = sigmoid( cos(params[0]) * cos( dot(x[b,:], W[0,:]) ) )
// Exact closed form of the reference circuit (product state + Z0 commutes with
// the CNOT chain). GEMM part done with V_WMMA_F32_16X16X4_F32.

typedef __attribute__((ext_vector_type(2))) float v2f;
typedef __attribute__((ext_vector_type(8))) float v8f;

#define FEAT 256

__global__ __launch_bounds__(256) void qcirc_wmma_kernel(
    const float* __restrict__ x,      // [B, 256]
    const float* __restrict__ W,      // [14, 256]
    const float* __restrict__ params, // [14]
    float* __restrict__ out)          // [B]
{
    const int lane = threadIdx.x & 31;
    const int wave = threadIdx.x >> 5;
    const int tile = blockIdx.x * 8 + wave; // 16-row batch tile per wave
    const int m0   = tile * 16;

    const int r     = lane & 15;          // A: row M ; B: col N (= W row)
    const int khalf = (lane >> 4) * 2;    // lanes 0-15: K+{0,1}; 16-31: K+{2,3}

    const float* xrow = x + (size_t)(m0 + r) * FEAT + khalf;

    // B-matrix: column n = W row n; zero-pad n = 14,15 without OOB reads.
    const bool  nvalid = (r < 14);
    const float wscale = nvalid ? 1.0f : 0.0f;
    const float* wrow  = W + (size_t)(nvalid ? r : 0) * FEAT + khalf;

    v8f acc = {0.f, 0.f, 0.f, 0.f, 0.f, 0.f, 0.f, 0.f};

    #pragma unroll 8
    for (int k0 = 0; k0 < FEAT; k0 += 4) {
        v2f a, b;
        a.x = xrow[k0 + 0];
        a.y = xrow[k0 + 1];
        b.x = wrow[k0 + 0] * wscale;
        b.y = wrow[k0 + 1] * wscale;
        // 8 args: (neg_a, A, neg_b, B, c_mod, C, reuse_a, reuse_b)
        acc = __builtin_amdgcn_wmma_f32_16x16x4_f32(
            false, a, false, b, (short)0, acc, false, false);
    }

    // Column N=0 of D: lane 0 holds rows m0+0..7 (acc[0..7]),
    //                  lane 16 holds rows m0+8..15 (acc[0..7]).
    const float cp0 = __cosf(params[0]);
    if ((lane & 15) == 0) {
        const int rowbase = m0 + (lane >> 4) * 8;
        #pragma unroll
        for (int i = 0; i < 8; ++i) {
            float z = cp0 * __cosf(acc[i]);
            out[rowbase + i] = 1.0f / (1.0f + __expf(-z));
        }
    }
}

extern "C" void kernel_launch(void* const* d_in, const int* in_sizes, int n_in,
                              void* d_out, int out_size, void* d_ws, size_t ws_size,
                              hipStream_t stream) {
    (void)n_in; (void)d_ws; (void)ws_size;
    const float* x      = (const float*)d_in[0];
    const float* W      = (const float*)d_in[1];
    const float* params = (const float*)d_in[2];
    float* out          = (float*)d_out;

    const int batch = out_size;              // 4096
    const int tiles = batch / 16;            // 256 waves
    const int blocks = tiles / 8;            // 8 waves (256 thr) per block
    hipLaunchKernelGGL(qcirc_wmma_kernel, dim3(blocks), dim3(256), 0, stream,
                       x, W, params, out);
}